// SignedGCN_17437567222139
// MI455X (gfx1250) — compile-verified
//
#include <hip/hip_runtime.h>
#include <hip/hip_bf16.h>

typedef float v2f __attribute__((ext_vector_type(2)));
typedef float v8f __attribute__((ext_vector_type(8)));

// ---------------------------------------------------------------------------
// Per-edge degree count: cnt[dst[e]] += 1
// ---------------------------------------------------------------------------
__global__ void count_kernel(const long long* __restrict__ dst,
                             float* __restrict__ cnt, int E) {
    int e = blockIdx.x * blockDim.x + threadIdx.x;
    if (e < E) atomicAdd(&cnt[(int)dst[e]], 1.0f);
}

// ---------------------------------------------------------------------------
// Scatter-add of 64-wide feature rows: sum[dst[e]] += feat[src[e], off:off+64]
// 16 threads per edge, float4 gathers (L2-resident tables on MI455X).
// ---------------------------------------------------------------------------
__global__ void scatter_add_kernel(const float* __restrict__ feat, int ld, int foff,
                                   const long long* __restrict__ src,
                                   const long long* __restrict__ dst,
                                   float* __restrict__ sum, int E) {
    int t = blockIdx.x * blockDim.x + threadIdx.x;
    int e = t >> 4;
    int c = t & 15;
    if (e >= E) return;
    long long s = src[e];
    long long d = dst[e];
    const float4 v = *(const float4*)(feat + (size_t)s * ld + foff + c * 4);
    float* p = sum + (size_t)d * 64 + c * 4;
    atomicAdd(p + 0, v.x);
    atomicAdd(p + 1, v.y);
    atomicAdd(p + 2, v.z);
    atomicAdd(p + 3, v.w);
}

// ---------------------------------------------------------------------------
// One 64-wide K segment of the WMMA GEMM: 16 x V_WMMA_F32_16X16X4_F32.
// row  : per-lane A-row pointer (A row m = lane&15 of the tile)
// invf : segment-mean normalizer folded into the A fragment
// wseg : per-lane B pointer = W + jn*K + seg*64 (B col n = lane&15)
// half : lane>>4 selects K pair {0,1} vs {2,3} per ISA f32 A/B layout
// ---------------------------------------------------------------------------
__device__ __forceinline__ v8f
wmma_seg64(v8f acc, const float* __restrict__ row, float invf,
           const float* __restrict__ wseg, int half) {
#pragma unroll 8
    for (int k = 0; k < 64; k += 4) {
        v2f a, b;
        a.x = row[k + 2 * half] * invf;
        a.y = row[k + 2 * half + 1] * invf;
        b.x = wseg[k + 2 * half];
        b.y = wseg[k + 2 * half + 1];
        acc = __builtin_amdgcn_wmma_f32_16x16x4_f32(
            /*neg_a=*/false, a, /*neg_b=*/false, b,
            /*c_mod=*/(short)0, acc, /*reuse_a=*/false, /*reuse_b=*/false);
    }
    return acc;
}

// ---------------------------------------------------------------------------
// Fused segment-mean-normalize + GEMM + bias + ReLU.
// Virtual A row for node m = [s0[m]/c0 | s1[m]/c1 (| s2[m]/c2)], 64 cols/seg.
// out[m, out_off+j] = relu( sum_k A(m,k) * W[j,k] + bias[j] ), W OUTxK row-major.
// Block = 128 threads (4 waves); wave w computes the 16x16 tile at cols w*16.
// NSEG is compile-time: K = NSEG*64 (2 -> layer 1, 3 -> layer 2).
// ---------------------------------------------------------------------------
template <int NSEG>
__global__ void __launch_bounds__(128)
wmma_gemm_relu(const float* __restrict__ s0, int ld0, int off0, const float* __restrict__ c0,
               const float* __restrict__ s1, int ld1, int off1, const float* __restrict__ c1,
               const float* __restrict__ s2, int ld2, int off2, const float* __restrict__ c2,
               const float* __restrict__ W, const float* __restrict__ bias,
               float* __restrict__ out, int out_ld, int out_off, int N) {
    constexpr int K = NSEG * 64;
    const int lane = threadIdx.x & 31;
    const int wave = threadIdx.x >> 5;
    const int m0   = blockIdx.x * 16;
    const int j0   = wave * 16;
    const int half = lane >> 4;      // 0: K pair {0,1}, 1: K pair {2,3}
    const int mr   = lane & 15;      // A row within tile / B column within tile

    int m = m0 + mr;
    if (m >= N) m = N - 1;           // clamp loads; stores masked below
    const int jn = j0 + mr;          // output column for this lane

    const float* r0 = s0 + (size_t)m * ld0 + off0;
    const float* r1 = s1 + (size_t)m * ld1 + off1;
    const float* r2 = s2 + (size_t)m * ld2 + off2;
    const float inv0 = c0 ? 1.0f / fmaxf(c0[m], 1.0f) : 1.0f;
    const float inv1 = c1 ? 1.0f / fmaxf(c1[m], 1.0f) : 1.0f;
    const float inv2 = (NSEG == 3 && c2) ? 1.0f / fmaxf(c2[m], 1.0f) : 1.0f;

    const float* w0 = W + (size_t)jn * K;

    v8f acc = {};
    acc = wmma_seg64(acc, r0, inv0, w0, half);
    acc = wmma_seg64(acc, r1, inv1, w0 + 64, half);
    if (NSEG == 3) acc = wmma_seg64(acc, r2, inv2, w0 + 128, half);

    const float bj = bias[jn];
#pragma unroll
    for (int r = 0; r < 8; ++r) {
        const int row = m0 + r + 8 * half;     // D layout: VGPR r -> M = r + 8*half
        if (row < N) {
            float v = acc[r] + bj;
            out[(size_t)row * out_ld + out_off + jn] = v > 0.0f ? v : 0.0f;
        }
    }
}

// ---------------------------------------------------------------------------
// Host-side orchestration (graph-capture safe: only async ops on `stream`).
// ---------------------------------------------------------------------------
extern "C" void kernel_launch(void* const* d_in, const int* in_sizes, int n_in,
                              void* d_out, int out_size, void* d_ws, size_t ws_size,
                              hipStream_t stream) {
    const float*     x   = (const float*)d_in[0];
    const long long* pei = (const long long*)d_in[1];   // (2, E) int64: row0=src, row1=dst
    const long long* nei = (const long long*)d_in[2];
    const float* Wp1 = (const float*)d_in[3];
    const float* bp1 = (const float*)d_in[4];
    const float* Wn1 = (const float*)d_in[5];
    const float* bn1 = (const float*)d_in[6];
    const float* Wp2 = (const float*)d_in[7];
    const float* bp2 = (const float*)d_in[8];
    const float* Wn2 = (const float*)d_in[9];
    const float* bn2 = (const float*)d_in[10];

    const int N = in_sizes[0] / 64;     // 100000
    const int E = in_sizes[1] / 2;      // 1600000

    // Workspace layout (floats): cnt_p[N] cnt_n[N] buf0..buf3[N*64] z[N*128]
    float* f     = (float*)d_ws;
    float* cnt_p = f;
    float* cnt_n = f + N;
    float* buf0  = f + (size_t)2 * N;
    float* buf1  = buf0 + (size_t)N * 64;
    float* buf2  = buf1 + (size_t)N * 64;
    float* buf3  = buf2 + (size_t)N * 64;
    float* z     = buf3 + (size_t)N * 64;
    float* out   = (float*)d_out;

    const int tb = 256;
    const int cgrid = (E + tb - 1) / tb;
    const long long tscatter = (long long)E * 16;
    const int sgrid = (int)((tscatter + tb - 1) / tb);
    const int ggrid = (N + 15) / 16;

    // Zero counts + all 4 sum buffers (z is fully overwritten by layer-1 GEMMs).
    hipMemsetAsync(f, 0, sizeof(float) * ((size_t)2 * N + (size_t)256 * N), stream);

    // Degrees (shared across both layers).
    count_kernel<<<cgrid, tb, 0, stream>>>(pei + E, cnt_p, E);
    count_kernel<<<cgrid, tb, 0, stream>>>(nei + E, cnt_n, E);

    // Layer 1 segment sums: mp = sum x[src] over pei, mn over nei.
    scatter_add_kernel<<<sgrid, tb, 0, stream>>>(x, 64, 0, pei, pei + E, buf0, E);
    scatter_add_kernel<<<sgrid, tb, 0, stream>>>(x, 64, 0, nei, nei + E, buf1, E);

    // Layer 1: z[:, :64] = relu([mp, x] @ Wp1^T + bp1); z[:, 64:] similarly.
    wmma_gemm_relu<2><<<ggrid, 128, 0, stream>>>(buf0, 64, 0, cnt_p,
                                                 x,    64, 0, nullptr,
                                                 x,    64, 0, nullptr,
                                                 Wp1, bp1, z, 128, 0,  N);
    wmma_gemm_relu<2><<<ggrid, 128, 0, stream>>>(buf1, 64, 0, cnt_n,
                                                 x,    64, 0, nullptr,
                                                 x,    64, 0, nullptr,
                                                 Wn1, bn1, z, 128, 64, N);

    // Re-zero sum buffers for layer 2.
    hipMemsetAsync(buf0, 0, sizeof(float) * (size_t)256 * N, stream);

    // Layer 2 segment sums: pp, pn, np_, nn.
    scatter_add_kernel<<<sgrid, tb, 0, stream>>>(z, 128, 0,  pei, pei + E, buf0, E); // pp  = sum zp over pei
    scatter_add_kernel<<<sgrid, tb, 0, stream>>>(z, 128, 64, nei, nei + E, buf1, E); // pn  = sum zn over nei
    scatter_add_kernel<<<sgrid, tb, 0, stream>>>(z, 128, 64, pei, pei + E, buf2, E); // np_ = sum zn over pei
    scatter_add_kernel<<<sgrid, tb, 0, stream>>>(z, 128, 0,  nei, nei + E, buf3, E); // nn  = sum zp over nei

    // Layer 2: out[:, :64] = relu([pp, pn, zp] @ Wp2^T + bp2); out[:, 64:] similarly.
    wmma_gemm_relu<3><<<ggrid, 128, 0, stream>>>(buf0, 64, 0, cnt_p,
                                                 buf1, 64, 0, cnt_n,
                                                 z,   128, 0, nullptr,
                                                 Wp2, bp2, out, 128, 0,  N);
    wmma_gemm_relu<3><<<ggrid, 128, 0, stream>>>(buf2, 64, 0, cnt_p,
                                                 buf3, 64, 0, cnt_n,
                                                 z,   128, 64, nullptr,
                                                 Wn2, bn2, out, 128, 64, N);
}